// SampleLoss_77300821393881
// MI455X (gfx1250) — compile-verified
//
#include <hip/hip_runtime.h>

#define HW        262144          // 512*512 pixels per image
#define NBATCH    32
#define NBINS     25
#define USED_BINS 24
#define CHUNKS    16              // blocks per image
#define CHUNK_V4  (HW / CHUNKS / 4)   // 4096 float4 per block
#define BLOCK     256             // 8 wave32s
#define COPIES    32              // per-lane LDS histogram copies
#define CSTR      33              // padded stride (bank spread)

typedef __attribute__((ext_vector_type(8))) float v8f;
typedef __attribute__((ext_vector_type(2))) float v2f;

// ---------------- workspace zeroing (counts live in d_ws) ----------------
__global__ void SampleLoss_zero_kernel(unsigned* __restrict__ cnt, int n) {
    int i = blockIdx.x * blockDim.x + threadIdx.x;
    if (i < n) cnt[i] = 0u;
}

// ---------------- streaming histogram pass ----------------
__device__ __forceinline__ void bin_add(unsigned* h, float v) {
    const float STEP = 10.2f;            // (255-0)/25 in f32, matches jnp edges
    const float INVS = 1.0f / 10.2f;
    int idx = (int)(v * INVS);           // v >= 0 -> trunc == floor
    // exact boundary correction against edges[i] = fl32(STEP * i)
    if (v < STEP * (float)idx)            idx -= 1;
    else if (v >= STEP * (float)(idx + 1)) idx += 1;
    if ((unsigned)idx < (unsigned)USED_BINS) atomicAdd(&h[idx], 1u);  // ds_add_u32
}

__global__ void __launch_bounds__(BLOCK)
SampleLoss_hist_kernel(const float* __restrict__ in, const float* __restrict__ tg,
                       unsigned* __restrict__ cnt) {
    __shared__ unsigned lh[COPIES * CSTR];
    const int tid = threadIdx.x;
    for (int i = tid; i < COPIES * CSTR; i += BLOCK) lh[i] = 0u;
    __syncthreads();

    const int img   = blockIdx.x >> 4;            // /CHUNKS ; 0..63
    const int chunk = blockIdx.x & (CHUNKS - 1);
    const int batch = img & (NBATCH - 1);
    const float* base = (img < NBATCH) ? in : tg;
    const float4* p = (const float4*)(base + (size_t)batch * HW) + (size_t)chunk * CHUNK_V4;

    unsigned* myh = &lh[(tid & 31) * CSTR];       // private copy per lane id
    for (int j = tid; j < CHUNK_V4; j += BLOCK) {
        float4 v = p[j];                          // global_load_b128
        bin_add(myh, v.x);
        bin_add(myh, v.y);
        bin_add(myh, v.z);
        bin_add(myh, v.w);
    }
    __syncthreads();

    if (tid < NBINS) {
        unsigned s = 0;
        #pragma unroll
        for (int c = 0; c < COPIES; ++c) s += lh[c * CSTR + tid];
        atomicAdd(&cnt[img * NBINS + tid], s);    // global_atomic_add_u32
    }
}

// ---------------- WMMA epilogue: batch reduction + loss ----------------
// D = ones(16x4) x B(4x16) + C   =>  D[m,n] = sum_k B[k,n] (all rows equal).
// Column n of B lives in lanes {n, n+16} (2 VGPRs each) = 4 K-slots; we put 4
// distinct batches there per iteration, so the accumulated column sum covers
// all 32 batches regardless of the exact K-slot ordering within the lane pair.
__global__ void SampleLoss_final_kernel(const unsigned* __restrict__ cnt,
                                        float* __restrict__ out) {
    const int lane = threadIdx.x;     // 1 wave32, EXEC all ones
    const int n = lane & 15;
    const int h = lane >> 4;
    const unsigned* cin = cnt;                    // [32][25]
    const unsigned* ctg = cnt + NBATCH * NBINS;   // [32][25]

    const v2f ones = {1.0f, 1.0f};
    v8f ad0 = {}, an0 = {}, at0 = {};             // tile0: bins 0..15
    v8f ad1 = {}, an1 = {}, at1 = {};             // tile1: bins 16..23 (rest 0)
    const int b2 = 16 + n;
    const bool ok = (b2 < USED_BINS);

    #pragma unroll
    for (int i = 0; i < 8; ++i) {
        const int bA = 4 * i + 2 * h;
        const int bB = bA + 1;
        // tile 0
        float ia = (float)cin[bA * NBINS + n];
        float ta = (float)ctg[bA * NBINS + n];
        float ib = (float)cin[bB * NBINS + n];
        float tb = (float)ctg[bB * NBINS + n];
        v2f bd = {fabsf(ia - ta), fabsf(ib - tb)};
        v2f bn = {ia, ib};
        v2f bt = {ta, tb};
        ad0 = __builtin_amdgcn_wmma_f32_16x16x4_f32(false, ones, false, bd, (short)0, ad0, false, false);
        an0 = __builtin_amdgcn_wmma_f32_16x16x4_f32(false, ones, false, bn, (short)0, an0, false, false);
        at0 = __builtin_amdgcn_wmma_f32_16x16x4_f32(false, ones, false, bt, (short)0, at0, false, false);
        // tile 1
        float ia1 = ok ? (float)cin[bA * NBINS + b2] : 0.0f;
        float ta1 = ok ? (float)ctg[bA * NBINS + b2] : 0.0f;
        float ib1 = ok ? (float)cin[bB * NBINS + b2] : 0.0f;
        float tb1 = ok ? (float)ctg[bB * NBINS + b2] : 0.0f;
        v2f bd1 = {fabsf(ia1 - ta1), fabsf(ib1 - tb1)};
        v2f bn1 = {ia1, ib1};
        v2f bt1 = {ta1, tb1};
        ad1 = __builtin_amdgcn_wmma_f32_16x16x4_f32(false, ones, false, bd1, (short)0, ad1, false, false);
        an1 = __builtin_amdgcn_wmma_f32_16x16x4_f32(false, ones, false, bn1, (short)0, an1, false, false);
        at1 = __builtin_amdgcn_wmma_f32_16x16x4_f32(false, ones, false, bt1, (short)0, at1, false, false);
    }

    // acc[0] holds row 0 (or identical row 8) of D: the per-column (per-bin) sums.
    const float P = (float)HW;
    float l0 = fminf(fmaxf(ad0[0] - 0.5f * fminf(an0[0], at0[0]), 0.0f), P) / P;
    float l1 = fminf(fmaxf(ad1[0] - 0.5f * fminf(an1[0], at1[0]), 0.0f), P) / P;
    // bins 0..15 from lanes 0..15 (tile0); bins 16..23 from lanes 0..7 (tile1)
    float c = ((lane < 16) ? l0 : 0.0f) + ((lane < 8) ? l1 : 0.0f);
    #pragma unroll
    for (int off = 16; off > 0; off >>= 1) c += __shfl_xor(c, off, 32);
    if (lane == 0) out[0] = c / (float)USED_BINS;
}

extern "C" void kernel_launch(void* const* d_in, const int* in_sizes, int n_in,
                              void* d_out, int out_size, void* d_ws, size_t ws_size,
                              hipStream_t stream) {
    const float* in  = (const float*)d_in[0];   // _input  (32,1,512,512) f32
    const float* tg  = (const float*)d_in[1];   // _target (32,1,512,512) f32
    unsigned* cnt = (unsigned*)d_ws;            // [2][32][25] u32 = 6400 B
    float* out = (float*)d_out;

    const int ncnt = 2 * NBATCH * NBINS;        // 1600
    SampleLoss_zero_kernel<<<(ncnt + 255) / 256, 256, 0, stream>>>(cnt, ncnt);
    SampleLoss_hist_kernel<<<2 * NBATCH * CHUNKS, BLOCK, 0, stream>>>(in, tg, cnt);
    SampleLoss_final_kernel<<<1, 32, 0, stream>>>(cnt, out);
}